// binGNN_68891275428348
// MI455X (gfx1250) — compile-verified
//
#include <hip/hip_runtime.h>
#include <hip/hip_bf16.h>

typedef __attribute__((ext_vector_type(2))) float v2f;
typedef __attribute__((ext_vector_type(8))) float v8f;

#define NEG_SLOPE 0.2f

// ---------------------------------------------------------------------------
// GEMM h[N,HC] = x[N,K] @ W[K,HC] using V_WMMA_F32_16X16X4_F32 (full fp32).
// One wave produces one 16x16 output tile; 4 waves per block share an LDS
// staging of the 16-wide W column slab (K*16 floats).
// A frag (16x4 f32): lanes 0-15 -> row m0+lane, VGPR0=K+0, VGPR1=K+1;
//                    lanes 16-31 -> row m0+lane-16, VGPR0=K+2, VGPR1=K+3.
// B frag (4x16 f32): lanes 0-15 -> col n0+lane, K+0/K+1; lanes 16-31 K+2/K+3.
// C/D (16x16 f32):   VGPR i: lanes 0-15 row m0+i, lanes 16-31 row m0+8+i.
// ---------------------------------------------------------------------------
__global__ __launch_bounds__(128)
void gat_gemm_wmma(const float* __restrict__ x, const float* __restrict__ W,
                   float* __restrict__ h, int N, int K, int HC) {
    extern __shared__ float Wlds[];                 // K*16 floats
    const int lane = threadIdx.x & 31;
    const int wave = threadIdx.x >> 5;
    const int n0   = blockIdx.y * 16;

    // cooperative stage of W[:, n0:n0+16] into LDS
    for (int i = threadIdx.x; i < K * 16; i += blockDim.x) {
        int kk = i >> 4, nn = i & 15;
        Wlds[i] = W[kk * HC + n0 + nn];
    }
    __syncthreads();

    const int m_tile = blockIdx.x * 4 + wave;       // wave-uniform
    const int m0 = m_tile * 16;
    if (m0 >= N) return;                            // whole wave exits together

    const int mrow = m0 + (lane & 15);
    const int koff = (lane >> 4) * 2;               // 0 for lanes 0-15, 2 for 16-31
    const int nsub = lane & 15;

    const float* xrow = x + (size_t)mrow * K;
    v8f acc = {};
    for (int k0 = 0; k0 < K; k0 += 4) {
        v2f a, b;
        a.x = xrow[k0 + koff];
        a.y = xrow[k0 + koff + 1];
        b.x = Wlds[(k0 + koff)     * 16 + nsub];
        b.y = Wlds[(k0 + koff + 1) * 16 + nsub];
        acc = __builtin_amdgcn_wmma_f32_16x16x4_f32(
            /*neg_a=*/false, a, /*neg_b=*/false, b,
            /*c_mod=*/(short)0, acc, /*reuse_a=*/false, /*reuse_b=*/false);
    }

    const int rbase = m0 + ((lane >> 4) << 3);      // +8 rows for upper half-wave
    const int ncol  = n0 + nsub;
#pragma unroll
    for (int i = 0; i < 8; ++i)
        h[(size_t)(rbase + i) * HC + ncol] = acc[i];
}

// ---------------------------------------------------------------------------
// Per-node attention pre-scores: es[n,h] = <h[n,h,:], a_src[h,:]>, same for ed.
// ---------------------------------------------------------------------------
__global__ void gat_scores(const float* __restrict__ h,
                           const float* __restrict__ a_src,
                           const float* __restrict__ a_dst,
                           float* __restrict__ es, float* __restrict__ ed,
                           int N, int H, int C) {
    int t = blockIdx.x * blockDim.x + threadIdx.x;
    if (t >= N * H) return;
    int n = t / H, hh = t - n * H;
    const float* hp = h + (size_t)n * H * C + hh * C;
    float s = 0.f, d = 0.f;
    for (int c = 0; c < C; ++c) {
        float v = hp[c];
        s += v * a_src[hh * C + c];
        d += v * a_dst[hh * C + c];
    }
    es[t] = s;
    ed[t] = d;
}

// ---------------------------------------------------------------------------
// Per-layer state init: m = -inf, denom = 0, accum = 0.
// ---------------------------------------------------------------------------
__global__ void gat_init(float* __restrict__ m, float* __restrict__ denom,
                         float* __restrict__ accum, int NH, int NHC) {
    int t = blockIdx.x * blockDim.x + threadIdx.x;
    if (t < NH) { m[t] = -3.0e38f; denom[t] = 0.f; }
    if (t < NHC) accum[t] = 0.f;
}

// ---------------------------------------------------------------------------
// Edge pass 1: segment max of leaky-relu attention logits by destination.
// Edge idx >= E encodes self-loop (src = dst = idx - E).
// ---------------------------------------------------------------------------
__global__ void edge_max(const int* __restrict__ ei, int E, int N,
                         const float* __restrict__ es, const float* __restrict__ ed,
                         float* __restrict__ m, int H) {
    int idx = blockIdx.x * blockDim.x + threadIdx.x;
    int Etot = E + N;
    if (idx >= Etot) return;
    int s = (idx < E) ? ei[idx]     : (idx - E);
    int d = (idx < E) ? ei[E + idx] : (idx - E);
    for (int hh = 0; hh < H; ++hh) {
        float v = es[s * H + hh] + ed[d * H + hh];
        v = (v > 0.f) ? v : NEG_SLOPE * v;
        atomicMax(&m[d * H + hh], v);               // global_atomic_max_num_f32
    }
}

// ---------------------------------------------------------------------------
// Edge pass 2 (fused softmax numerator): ex = exp(e - m[dst]);
// denom[dst] += ex; accum[dst,:] += h[src,:] * ex.
// ---------------------------------------------------------------------------
__global__ void edge_agg(const int* __restrict__ ei, int E, int N,
                         const float* __restrict__ es, const float* __restrict__ ed,
                         const float* __restrict__ m, const float* __restrict__ h,
                         float* __restrict__ denom, float* __restrict__ accum,
                         int H, int C) {
    int idx = blockIdx.x * blockDim.x + threadIdx.x;
    int Etot = E + N;
    if (idx >= Etot) return;
    int s = (idx < E) ? ei[idx]     : (idx - E);
    int d = (idx < E) ? ei[E + idx] : (idx - E);
    int HC = H * C;
    for (int hh = 0; hh < H; ++hh) {
        float v = es[s * H + hh] + ed[d * H + hh];
        v = (v > 0.f) ? v : NEG_SLOPE * v;
        float ex = __expf(v - m[d * H + hh]);
        atomicAdd(&denom[d * H + hh], ex);
        const float* hs = h + (size_t)s * HC + hh * C;
        float* ap = accum + (size_t)d * HC + hh * C;
        for (int c = 0; c < C; ++c)
            atomicAdd(&ap[c], hs[c] * ex);
    }
}

// ---------------------------------------------------------------------------
// Node finalize: out = relu(accum / denom + bias).
// ---------------------------------------------------------------------------
__global__ void node_finalize(const float* __restrict__ accum,
                              const float* __restrict__ denom,
                              const float* __restrict__ b,
                              float* __restrict__ out, int N, int H, int C) {
    int HC = H * C;
    int t = blockIdx.x * blockDim.x + threadIdx.x;
    if (t >= N * HC) return;
    int n = t / HC, j = t - n * HC, hh = j / C;
    float v = accum[t] / denom[n * H + hh] + b[j];
    out[t] = (v > 0.f) ? v : 0.f;
}

// ---------------------------------------------------------------------------
// Graph mean-pool + FC + sigmoid.
// ---------------------------------------------------------------------------
__global__ void pool_init(float* __restrict__ sums, float* __restrict__ cnt) {
    int t = blockIdx.x * blockDim.x + threadIdx.x;
    if (t < 64 * 16) sums[t] = 0.f;
    if (t < 64) cnt[t] = 0.f;
}

__global__ void pool_acc(const float* __restrict__ h, const int* __restrict__ batch,
                         float* __restrict__ sums, float* __restrict__ cnt, int N) {
    int n = blockIdx.x * blockDim.x + threadIdx.x;
    if (n >= N) return;
    int g = batch[n];
    for (int j = 0; j < 16; ++j)
        atomicAdd(&sums[g * 16 + j], h[(size_t)n * 16 + j]);
    atomicAdd(&cnt[g], 1.0f);
}

__global__ void pool_fc(const float* __restrict__ sums, const float* __restrict__ cnt,
                        const float* __restrict__ fcW, const float* __restrict__ fcb,
                        float* __restrict__ out) {
    int g = blockIdx.x * blockDim.x + threadIdx.x;
    if (g >= 64) return;
    float c = fmaxf(cnt[g], 1.0f);
    float acc = fcb[0];
    for (int j = 0; j < 16; ++j)
        acc += (sums[g * 16 + j] / c) * fcW[j];
    out[g] = 1.f / (1.f + __expf(-acc));
}

// ---------------------------------------------------------------------------
extern "C" void kernel_launch(void* const* d_in, const int* in_sizes, int n_in,
                              void* d_out, int out_size, void* d_ws, size_t ws_size,
                              hipStream_t stream) {
    const float* x     = (const float*)d_in[0];
    const int*   ei    = (const int*)d_in[1];
    const int*   batch = (const int*)d_in[2];
    const float* fcW   = (const float*)d_in[23];
    const float* fcb   = (const float*)d_in[24];

    const int N = in_sizes[0] / 128;     // 50000
    const int E = in_sizes[1] / 2;       // 1600000
    const int Etot = E + N;              // with self-loops

    // workspace carve-up (fp32), 256B-aligned chunks
    float* ws = (float*)d_ws;
    size_t off = 0;
    auto carve = [&](size_t n) { float* p = ws + off; off += (n + 63) & ~(size_t)63; return p; };
    float* h     = carve((size_t)N * 128);
    float* accum = carve((size_t)N * 128);
    float* bufA  = carve((size_t)N * 128);
    float* bufB  = carve((size_t)N * 128);
    float* es    = carve((size_t)N * 2);
    float* ed    = carve((size_t)N * 2);
    float* mbuf  = carve((size_t)N * 2);
    float* denom = carve((size_t)N * 2);
    float* sums  = carve(64 * 16);
    float* cnt   = carve(64);

    static const int FIN[5] = {128, 32, 64, 128, 64};
    static const int HH[5]  = {1, 2, 2, 2, 2};
    static const int CC[5]  = {32, 32, 64, 32, 8};

    const float* cur = x;
    float* nxt = bufA;

    for (int li = 0; li < 5; ++li) {
        const float* W   = (const float*)d_in[3 + 4 * li];
        const float* asc = (const float*)d_in[4 + 4 * li];
        const float* adc = (const float*)d_in[5 + 4 * li];
        const float* bb  = (const float*)d_in[6 + 4 * li];
        const int K = FIN[li], Hn = HH[li], C = CC[li], HC = Hn * C;
        const int NH = N * Hn, NHC = N * HC;

        gat_init<<<(NHC + 255) / 256, 256, 0, stream>>>(mbuf, denom, accum, NH, NHC);

        dim3 grid((N / 16 + 3) / 4, HC / 16);
        gat_gemm_wmma<<<grid, 128, K * 16 * sizeof(float), stream>>>(cur, W, h, N, K, HC);

        gat_scores<<<(NH + 255) / 256, 256, 0, stream>>>(h, asc, adc, es, ed, N, Hn, C);

        edge_max<<<(Etot + 255) / 256, 256, 0, stream>>>(ei, E, N, es, ed, mbuf, Hn);

        edge_agg<<<(Etot + 255) / 256, 256, 0, stream>>>(ei, E, N, es, ed, mbuf, h,
                                                         denom, accum, Hn, C);

        node_finalize<<<(NHC + 255) / 256, 256, 0, stream>>>(accum, denom, bb, nxt, N, Hn, C);

        cur = nxt;
        nxt = (nxt == bufA) ? bufB : bufA;
    }

    pool_init<<<5, 256, 0, stream>>>(sums, cnt);
    pool_acc<<<(N + 255) / 256, 256, 0, stream>>>(cur, batch, sums, cnt, N);
    pool_fc<<<1, 64, 0, stream>>>(sums, cnt, fcW, fcb, (float*)d_out);
}